// Attn_loc_90795608637907
// MI455X (gfx1250) — compile-verified
//
#include <hip/hip_runtime.h>
#include <stdint.h>

// Reference: out[i,j] = softmax_j( 1 / poi[current[i], history[j]] )
//   history: 4096 int32, current: 2048 int32, poi: 10000x10000 f32, out: 2048x4096 f32
//
// Strategy (MI455X): one workgroup per output row. TDM (tensor_load_to_lds)
// DMAs the needed 40KB matrix row into LDS as one contiguous 1-D tile; the
// random history-column gathers then hit the 64-bank LDS instead of issuing
// 4096 scattered global requests per row. The DMA is overlapped with the
// history index loads. Softmax reductions use wave32 shuffles + tiny LDS
// cross-wave arrays. Output is streamed with non-temporal stores.

#define N_POI     10000
#define SEQ_LEN   4096
#define STATE_LEN 2048
#define THREADS   256
#define NVEC      4          // 4 x float4 = 16 elements per thread

typedef unsigned int v4u __attribute__((ext_vector_type(4)));
typedef int          v8i __attribute__((ext_vector_type(8)));
typedef int          v4i __attribute__((ext_vector_type(4)));
typedef float        v4f __attribute__((ext_vector_type(4)));

__global__ __launch_bounds__(THREADS)
void attn_loc_softmax_kernel(const int* __restrict__ history,
                             const int* __restrict__ current,
                             const float* __restrict__ poi,
                             float* __restrict__ out)
{
    __shared__ __align__(16) float rowbuf[N_POI];   // 40000 B staged row
    __shared__ float redm[THREADS / 32];
    __shared__ float reds[THREADS / 32];

    const int row  = blockIdx.x;
    const int tid  = threadIdx.x;
    const int wid  = tid >> 5;
    const int lane = tid & 31;

    // Row index for this block (uniform scalar load).
    const int c = current[row];

    // Global byte address of the start of matrix row c.
    const uint64_t gaddr =
        (uint64_t)(uintptr_t)poi + (uint64_t)c * (uint64_t)(N_POI * 4);

    // LDS byte offset of rowbuf (addrspace(3) pointer value == LDS offset).
    const unsigned lds_base =
        (unsigned)(uintptr_t)(__attribute__((address_space(3))) float*)&rowbuf[0];

    // ---- Issue the TDM row DMA (wave 0 only; tensor ops are per-wave). ----
    if (tid < 32) {
        v4u g0;
        g0.x = 1u;                                   // count=1, is_restore=0, no gather
        g0.y = lds_base;                             // lds_addr (bytes)
        g0.z = (unsigned)(gaddr & 0xFFFFFFFFull);    // global_addr[31:0]
        g0.w = (unsigned)((gaddr >> 32) & 0x01FFFFFFull) | (2u << 30); // addr[56:32] | type=2

        v8i g1;
        g1[0] = (int)(2u << 16);                         // wg_mask=0, data_size=2 (4B)
        g1[1] = (int)((unsigned)(N_POI & 0xFFFF) << 16); // tensor_dim0[15:0]
        g1[2] = (int)((unsigned)(N_POI >> 16) | (1u << 16)); // dim0 hi | tensor_dim1=1 lo
        g1[3] = (int)((unsigned)N_POI << 16);            // tensor_dim1 hi=0 | tile_dim0=10000
        g1[4] = 1;                                       // tile_dim1=1, tile_dim2=0
        g1[5] = N_POI;                                   // tensor_dim0_stride[31:0]
        g1[6] = (int)((unsigned)(N_POI & 0xFFFF) << 16); // stride hi=0 | dim1_stride lo
        g1[7] = 0;                                       // dim1_stride hi

        v4i g2 = {0, 0, 0, 0};
        v4i g3 = {0, 0, 0, 0};
        v8i g4 = {0, 0, 0, 0, 0, 0, 0, 0};               // extra group (unused)

        __builtin_amdgcn_tensor_load_to_lds(g0, g1, g2, g3, g4, /*cpol=*/0);
    }

    // ---- Overlap: fetch this thread's 16 history indices while TDM runs. ----
    const v4i* h4 = (const v4i*)history;
    v4i h[NVEC];
#pragma unroll
    for (int k = 0; k < NVEC; ++k)
        h[k] = h4[tid + k * THREADS];

    // ---- Drain the DMA (issuing wave), then make LDS visible block-wide. ----
    if (tid < 32)
        __builtin_amdgcn_s_wait_tensorcnt(0);
    __syncthreads();

    // ---- Gather + reciprocal; track row max. ----
    v4f ev[NVEC];
    float m = -3.402823466e+38f;
#pragma unroll
    for (int k = 0; k < NVEC; ++k) {
        v4f e;
        e.x = 1.0f / rowbuf[h[k].x];
        e.y = 1.0f / rowbuf[h[k].y];
        e.z = 1.0f / rowbuf[h[k].z];
        e.w = 1.0f / rowbuf[h[k].w];
        ev[k] = e;
        m = fmaxf(m, fmaxf(fmaxf(e.x, e.y), fmaxf(e.z, e.w)));
    }

    // Wave32 max reduction, then cross-wave via LDS.
#pragma unroll
    for (int off = 16; off > 0; off >>= 1)
        m = fmaxf(m, __shfl_xor(m, off, 32));
    if (lane == 0) redm[wid] = m;
    __syncthreads();
    float bm = redm[0];
#pragma unroll
    for (int i = 1; i < THREADS / 32; ++i) bm = fmaxf(bm, redm[i]);

    // ---- exp(e - max), accumulate sum. ----
    float s = 0.0f;
#pragma unroll
    for (int k = 0; k < NVEC; ++k) {
        v4f e = ev[k];
        e.x = __expf(e.x - bm);
        e.y = __expf(e.y - bm);
        e.z = __expf(e.z - bm);
        e.w = __expf(e.w - bm);
        ev[k] = e;
        s += (e.x + e.y) + (e.z + e.w);
    }
#pragma unroll
    for (int off = 16; off > 0; off >>= 1)
        s += __shfl_xor(s, off, 32);
    if (lane == 0) reds[wid] = s;
    __syncthreads();
    float bs = 0.0f;
#pragma unroll
    for (int i = 0; i < THREADS / 32; ++i) bs += reds[i];
    const float inv = 1.0f / bs;

    // ---- Normalize and stream out (non-temporal coalesced 128b stores). ----
    v4f* o4 = (v4f*)(out + (size_t)row * SEQ_LEN);
#pragma unroll
    for (int k = 0; k < NVEC; ++k) {
        v4f p = ev[k];
        p *= inv;
        __builtin_nontemporal_store(p, &o4[tid + k * THREADS]);
    }
}

extern "C" void kernel_launch(void* const* d_in, const int* in_sizes, int n_in,
                              void* d_out, int out_size, void* d_ws, size_t ws_size,
                              hipStream_t stream) {
    (void)in_sizes; (void)n_in; (void)out_size; (void)d_ws; (void)ws_size;
    const int*   history = (const int*)d_in[0];
    const int*   current = (const int*)d_in[1];
    const float* poi     = (const float*)d_in[2];
    float*       out     = (float*)d_out;
    attn_loc_softmax_kernel<<<STATE_LEN, THREADS, 0, stream>>>(history, current, poi, out);
}